// GatedGCN_2d_77343771066510
// MI455X (gfx1250) — compile-verified
//
#include <hip/hip_runtime.h>
#include <math.h>

#define NN 100000
#define EE 1600000
#define ND 6400000LL   // N*D

typedef float v2f __attribute__((ext_vector_type(2)));
typedef float v8f __attribute__((ext_vector_type(8)));

// workspace layout (float offsets)
#define OFF_A1   (0LL*ND)
#define OFF_A2   (1LL*ND)
#define OFF_A3   (2LL*ND)
#define OFF_B1   (3LL*ND)
#define OFF_B2   (4LL*ND)
#define OFF_C1   (5LL*ND)
#define OFF_C2   (6LL*ND)
#define OFF_NUMF (7LL*ND)
#define OFF_DENF (8LL*ND)
#define OFF_NUMB (9LL*ND)
#define OFF_DENB (10LL*ND)
#define OFF_STATS (11LL*ND)   // [0:64) colsum [64:128) colsumsq [128:192) scale [192:256) shift

__global__ void zero_f32(float* __restrict__ p, long long n) {
    long long i = (long long)blockIdx.x * blockDim.x + threadIdx.x;
    if (i < n) p[i] = 0.0f;
}

// ---- 16x16 output tile of (X[16x64] @ W[64x64]) via 16 chained f32 WMMAs ----
__device__ __forceinline__ v8f tile_gemm_16x16x64(const float* __restrict__ X,  // 16 rows, stride 64
                                                  const float* __restrict__ W,  // 64x64 row-major
                                                  int n0, int lane) {
    const int lo = lane & 15;
    const int hi = lane >> 4;
    v8f c = {0.f, 0.f, 0.f, 0.f, 0.f, 0.f, 0.f, 0.f};
#pragma unroll
    for (int ks = 0; ks < 16; ++ks) {
        const int k = ks * 4 + 2 * hi;
        v2f a, b;
        a.x = X[lo * 64 + k];
        a.y = X[lo * 64 + k + 1];
        b.x = W[k * 64 + n0 + lo];
        b.y = W[(k + 1) * 64 + n0 + lo];
        c = __builtin_amdgcn_wmma_f32_16x16x4_f32(false, a, false, b, (short)0, c, false, false);
    }
    return c;
}

// ---- 7 node projections: h @ W + b ----
__global__ __launch_bounds__(256) void node_proj_kernel(
    const float* __restrict__ h,
    const float* __restrict__ W_A, const float* __restrict__ b_A,
    const float* __restrict__ W_B, const float* __restrict__ b_B,
    const float* __restrict__ W_C, const float* __restrict__ b_C,
    float* __restrict__ ws)
{
    const int lane = threadIdx.x & 31;
    const int wave = threadIdx.x >> 5;
    const int rt = blockIdx.x * 8 + wave;          // 16-row tile index
    if (rt * 16 >= NN) return;
    const int m  = blockIdx.y >> 2;                // matrix 0..6
    const int n0 = (blockIdx.y & 3) * 16;

    const float* W; const float* bias; float* out;
    switch (m) {
        case 0:  W = W_A;        bias = b_A;       out = ws + OFF_A1; break;
        case 1:  W = W_A + 4096; bias = b_A + 64;  out = ws + OFF_A2; break;
        case 2:  W = W_A + 8192; bias = b_A + 128; out = ws + OFF_A3; break;
        case 3:  W = W_B;        bias = b_B;       out = ws + OFF_B1; break;
        case 4:  W = W_B + 4096; bias = b_B + 64;  out = ws + OFF_B2; break;
        case 5:  W = W_C;        bias = b_C;       out = ws + OFF_C1; break;
        default: W = W_C + 4096; bias = b_C + 64;  out = ws + OFF_C2; break;
    }

    const int lo = lane & 15;
    const int hi = lane >> 4;
    const float* X = h + (long long)rt * 16 * 64;
    v8f c = tile_gemm_16x16x64(X, W, n0, lane);

    const float bv = bias[n0 + lo];
    float* orow = out + (long long)rt * 16 * 64 + n0 + lo;
#pragma unroll
    for (int i = 0; i < 8; ++i)
        orow[(i + 8 * hi) * 64] = c[i] + bv;
}

// ---- fused edge kernel: gate = sigmoid(P1[i1] + P2[i2] + efeat@W + b);
//      num[i2] += gate * Amat[i1];  den[i2] += gate ----
__global__ __launch_bounds__(128) void edge_kernel(
    const float* __restrict__ efeat, const float* __restrict__ W, const float* __restrict__ bias,
    const float* __restrict__ P1, const float* __restrict__ P2, const float* __restrict__ Amat,
    const int* __restrict__ idx1, const int* __restrict__ idx2,
    float* __restrict__ num, float* __restrict__ den)
{
    __shared__ int s1[16], s2[16];
    const long long e0 = (long long)blockIdx.x * 16;
    if (threadIdx.x < 16) {
        s1[threadIdx.x] = idx1[e0 + threadIdx.x];
        s2[threadIdx.x] = idx2[e0 + threadIdx.x];
    }
    __syncthreads();

    const int lane = threadIdx.x & 31;
    const int wave = threadIdx.x >> 5;
    const int n0 = wave * 16;
    const int lo = lane & 15;
    const int hi = lane >> 4;

    const float* X = efeat + e0 * 64;
    v8f c = tile_gemm_16x16x64(X, W, n0, lane);

    const int col = n0 + lo;
    const float bv = bias[col];
#pragma unroll
    for (int i = 0; i < 8; ++i) {
        const int m = i + 8 * hi;
        const long long r1 = (long long)s1[m] * 64 + col;
        const long long r2 = (long long)s2[m] * 64 + col;
        const float x = c[i] + bv + P1[r1] + P2[r2];
        const float g = 1.0f / (1.0f + __expf(-x));
        unsafeAtomicAdd(&num[r2], g * Amat[r1]);
        unsafeAtomicAdd(&den[r2], g);
    }
}

// ---- h_pre = A1h + num_f/(den_f+eps) + num_b/(den_b+eps); accumulate BN stats ----
__global__ __launch_bounds__(256) void combine_kernel(float* __restrict__ ws, float* __restrict__ hpre)
{
    __shared__ float ssum[256], ssq[256];
    const int col  = threadIdx.x & 63;
    const int rsub = threadIdx.x >> 6;
    const int r0 = blockIdx.x * 64;
    const float* A1 = ws + OFF_A1;
    const float* nf = ws + OFF_NUMF; const float* df = ws + OFF_DENF;
    const float* nb = ws + OFF_NUMB; const float* db = ws + OFF_DENB;

    float lsum = 0.f, lsq = 0.f;
#pragma unroll
    for (int i = 0; i < 16; ++i) {
        const int r = r0 + rsub + 4 * i;
        if (r < NN) {
            const long long o = (long long)r * 64 + col;
            const float v = A1[o] + nf[o] / (df[o] + 1e-6f) + nb[o] / (db[o] + 1e-6f);
            hpre[o] = v;
            lsum += v;
            lsq  += v * v;
        }
    }
    ssum[threadIdx.x] = lsum;
    ssq[threadIdx.x]  = lsq;
    __syncthreads();
    if (threadIdx.x < 64) {
        const float s = ssum[threadIdx.x] + ssum[threadIdx.x + 64] + ssum[threadIdx.x + 128] + ssum[threadIdx.x + 192];
        const float q = ssq[threadIdx.x]  + ssq[threadIdx.x + 64]  + ssq[threadIdx.x + 128]  + ssq[threadIdx.x + 192];
        float* stats = ws + OFF_STATS;
        unsafeAtomicAdd(&stats[threadIdx.x], s);
        unsafeAtomicAdd(&stats[64 + threadIdx.x], q);
    }
}

__global__ void finalize_bn(float* __restrict__ ws, const float* __restrict__ gamma, const float* __restrict__ beta)
{
    const int t = threadIdx.x;
    if (t < 64) {
        float* stats = ws + OFF_STATS;
        const float invN = 1.0f / (float)NN;
        const float mean = stats[t] * invN;
        float var = stats[64 + t] * invN - mean * mean;
        var = fmaxf(var, 0.0f);
        const float sc = gamma[t] * rsqrtf(var + 1e-5f);
        stats[128 + t] = sc;
        stats[192 + t] = beta[t] - mean * sc;
    }
}

__global__ __launch_bounds__(256) void apply_bn_kernel(float* __restrict__ hout,
                                                       const float* __restrict__ h,
                                                       const float* __restrict__ ws)
{
    const long long i = (long long)blockIdx.x * 256 + threadIdx.x;
    const int col = (int)(i & 63);
    const float* stats = ws + OFF_STATS;
    const float v = hout[i] * stats[128 + col] + stats[192 + col];
    hout[i] = fmaxf(v, 0.0f) + h[i];
}

extern "C" void kernel_launch(void* const* d_in, const int* in_sizes, int n_in,
                              void* d_out, int out_size, void* d_ws, size_t ws_size,
                              hipStream_t stream) {
    (void)in_sizes; (void)n_in; (void)out_size; (void)ws_size;
    const float* h   = (const float*)d_in[0];
    const float* e_f = (const float*)d_in[1];
    const float* e_b = (const float*)d_in[2];
    const int*   src = (const int*)d_in[3];
    const int*   dst = (const int*)d_in[4];
    const float* W_A = (const float*)d_in[5];
    const float* b_A = (const float*)d_in[6];
    const float* W_B = (const float*)d_in[7];
    const float* b_B = (const float*)d_in[8];
    const float* W_C = (const float*)d_in[9];
    const float* b_C = (const float*)d_in[10];
    const float* gam = (const float*)d_in[11];
    const float* bet = (const float*)d_in[12];
    float* out = (float*)d_out;
    float* ws  = (float*)d_ws;

    // zero segment-sum accumulators + BN stats
    {
        const long long n = 4LL * ND + 256;
        const int blocks = (int)((n + 255) / 256);
        zero_f32<<<blocks, 256, 0, stream>>>(ws + OFF_NUMF, n);
    }

    // 7 node projections (WMMA)
    node_proj_kernel<<<dim3((6250 + 7) / 8, 28), 256, 0, stream>>>(
        h, W_A, b_A, W_B, b_B, W_C, b_C, ws);

    // forward: gate = sigmoid(B1h[src] + B2h[dst] + e_f@W_B2 + b); reduce at dst
    edge_kernel<<<EE / 16, 128, 0, stream>>>(
        e_f, W_B + 8192, b_B + 128,
        ws + OFF_B1, ws + OFF_B2, ws + OFF_A2,
        src, dst, ws + OFF_NUMF, ws + OFF_DENF);

    // backward: gate = sigmoid(C1h[dst] + C2h[src] + e_b@W_C2 + b); reduce at src
    edge_kernel<<<EE / 16, 128, 0, stream>>>(
        e_b, W_C + 8192, b_C + 128,
        ws + OFF_C1, ws + OFF_C2, ws + OFF_A3,
        dst, src, ws + OFF_NUMB, ws + OFF_DENB);

    // combine + column stats
    combine_kernel<<<(NN + 63) / 64, 256, 0, stream>>>(ws, out);
    finalize_bn<<<1, 64, 0, stream>>>(ws, gam, bet);
    apply_bn_kernel<<<(int)(ND / 256), 256, 0, stream>>>(out, h, ws);

    // pass-through outputs e_f, e_b
    hipMemcpyAsync(out + ND, e_f, (size_t)EE * 64 * sizeof(float),
                   hipMemcpyDeviceToDevice, stream);
    hipMemcpyAsync(out + ND + (long long)EE * 64, e_b, (size_t)EE * 64 * sizeof(float),
                   hipMemcpyDeviceToDevice, stream);
}